// RelationGraphConv_8735963480237
// MI455X (gfx1250) — compile-verified
//
#include <hip/hip_runtime.h>

#define C 128          // C_IN == C_OUT
#define E 4            // NUM_EDGES
#define NB 8           // batch
#define NN 1024        // nodes

typedef float v2f __attribute__((ext_vector_type(2)));
typedef float v8f __attribute__((ext_vector_type(8)));

__device__ __forceinline__ v8f wmma_f32(v2f a, v2f b, v8f c) {
  // D = A(16x4) * B(4x16) + C ; 8 args: (neg_a, A, neg_b, B, c_mod, C, reuse_a, reuse_b)
  return __builtin_amdgcn_wmma_f32_16x16x4_f32(false, a, false, b, (short)0, c, false, false);
}

// ---------------------------------------------------------------------------
// Kernel 1: fused LayerNorm + dual GEMM  (xn @ [W_hs | W_hr])
//   hs  -> d_out   [b, n, c]
//   hr_t-> d_ws    [b, m, c, e]   (transposed so kernel2's B-fragment is one b64)
// Each workgroup: 16 rows of (b,n), all 640 output columns.
// Per wave: 1 hs tile (ld=128, compile-time) + 4 hr tiles (ld=512, compile-time)
// so all weight loads are immediate-offset clauses off a single base pointer.
// ---------------------------------------------------------------------------
__global__ __launch_bounds__(256) void ln_gemm_kernel(
    const float* __restrict__ x, const float* __restrict__ gamma,
    const float* __restrict__ beta,
    const float* __restrict__ W_hs, const float* __restrict__ b_hs,
    const float* __restrict__ W_hr, const float* __restrict__ b_hr,
    float* __restrict__ hs_out, float* __restrict__ hr_t) {
  __shared__ float xs[16 * 132];               // 16 rows, padded stride 132 (bank-safe)
  const int tid = threadIdx.x;
  const int rb  = blockIdx.x;                  // 0..511
  const int b   = rb >> 6;
  const int m0  = (rb & 63) << 4;

  // ---- LayerNorm: 16 lanes per row, 8 elems per lane ----
  const int row = tid >> 4;                    // 0..15
  const int cg  = (tid & 15) << 3;             // 0..120
  const float* xr = x + ((size_t)(b * NN + m0 + row)) * C + cg;
  float v[8];
  float s = 0.f, sq = 0.f;
#pragma unroll
  for (int i = 0; i < 8; ++i) { v[i] = xr[i]; s += v[i]; sq += v[i] * v[i]; }
#pragma unroll
  for (int off = 8; off >= 1; off >>= 1) {     // reduce within 16-lane row group
    s  += __shfl_xor(s,  off, 32);
    sq += __shfl_xor(sq, off, 32);
  }
  const float mu   = s * (1.f / 128.f);
  const float var  = sq * (1.f / 128.f) - mu * mu;
  const float rsig = rsqrtf(var + 1e-5f);
  float* xrow = xs + row * 132 + cg;
#pragma unroll
  for (int i = 0; i < 8; ++i)
    xrow[i] = (v[i] - mu) * rsig * gamma[cg + i] + beta[cg + i];
  __syncthreads();

  // ---- GEMM phase: A = xn[16 x 128] from LDS, B = weight tiles from global ----
  const int lane = tid & 31, wave = tid >> 5;
  const int lo = lane & 15, hi = lane >> 4, hi2 = hi << 1;
  const float* aA = xs + lo * 132 + hi2;       // + k gives A[lo, k+hi2 .. +1]
  const size_t rowbase = (size_t)(b * NN + m0 + hi * 8);

  // -- hs tile: columns [wave*16, wave*16+16) of W_hs --
  {
    const int col0 = wave << 4;
    const float* wq = W_hs + hi2 * C + col0 + lo;        // B[k+hi2, col0+lo]
    v8f acc = {};
#pragma unroll 4
    for (int k = 0; k < C; k += 4) {
      v2f a = *(const v2f*)(aA + k);
      v2f bf;
      bf.x = wq[k * C];                                  // immediate offsets
      bf.y = wq[(k + 1) * C];
      acc = wmma_f32(a, bf, acc);
    }
    const float bb = b_hs[col0 + lo];
    float* op = hs_out + rowbase * C + col0 + lo;
#pragma unroll
    for (int j = 0; j < 8; ++j) op[j * C] = acc[j] + bb;
  }

  // -- 4 hr tiles: columns [wave*64, wave*64+64) of W_hr (512 wide) --
#pragma unroll 1
  for (int i = 0; i < 4; ++i) {
    const int jt = (wave * 4 + i) << 4;                  // tile start column in W_hr
    const float* wq = W_hr + hi2 * (C * E) + jt + lo;
    v8f acc = {};
#pragma unroll 4
    for (int k = 0; k < C; k += 4) {
      v2f a = *(const v2f*)(aA + k);
      v2f bf;
      bf.x = wq[k * (C * E)];                            // immediate offsets (<=260KB)
      bf.y = wq[(k + 1) * (C * E)];
      acc = wmma_f32(a, bf, acc);
    }
    const int jc = jt + lo;                              // reference col j = e*C_OUT + c
    const float bb = b_hr[jc];
    const int e = jc >> 7, cc = jc & 127;
    float* op = hr_t + (rowbase * C + cc) * E + e;
#pragma unroll
    for (int j = 0; j < 8; ++j) op[j * (C * E)] = acc[j] + bb;
  }
}

// ---------------------------------------------------------------------------
// Kernel 2: fused relational aggregate + neighbour-count + normalize + add hs
//   out[b,n,c] = hs[b,n,c] + (sum_{m,e} adj[b,m,n,e]*hr_t[b,m,c,e]) / max(nn,1e-5)
//   nn[b,n]    = sum_{m,e} adj[b,m,n,e]   (fused: adjacency read exactly once)
// WG = 16n x 128c tile (8 waves, one 16x16 f32 acc each); K-loop over m, K=4 edges
// per WMMA. A and B fragments are each ONE coalesced global_load_b64 per step;
// unroll 8 -> 16 loads in flight per wave per iteration (two s_clauses).
// ---------------------------------------------------------------------------
__global__ __launch_bounds__(256) void einsum_kernel(
    const float* __restrict__ adj, const float* __restrict__ hr_t,
    float* __restrict__ out) {
  const int lane = threadIdx.x & 31, wave = threadIdx.x >> 5;
  const int lo = lane & 15, hi = lane >> 4;
  const int b  = blockIdx.y;
  const int n0 = blockIdx.x << 4;
  const int c0 = wave << 4;

  // A frag: adj[b, m, n0+lo, hi*2 .. +1]   B frag: hr_t[b, m, c0+lo, hi*2 .. +1]
  const v2f* pA = (const v2f*)(adj  + (size_t)b * NN * NN * E + (size_t)(n0 + lo) * E + hi * 2);
  const v2f* pB = (const v2f*)(hr_t + (size_t)b * NN * C  * E + (size_t)(c0 + lo) * E + hi * 2);
  const size_t sA = (size_t)NN * E / 2;   // 2048 v2f per m (16 KB)
  const size_t sB = (size_t)C  * E / 2;   // 256  v2f per m (2 KB)

  v8f acc = {};
  float ns = 0.f;                          // per-lane partial of num_neighbours
#pragma unroll 1
  for (int m = 0; m < NN; m += 8) {
    v2f a[8], bf[8];
#pragma unroll
    for (int u = 0; u < 8; ++u)            // adjacency: streamed once -> non-temporal
      a[u] = __builtin_nontemporal_load(pA + (size_t)u * sA);
#pragma unroll
    for (int u = 0; u < 8; ++u)            // hr_t: reused by 64 WGs -> keep cacheable
      bf[u] = pB[(size_t)u * sB];
    pA += 8 * sA; pB += 8 * sB;
#pragma unroll
    for (int u = 0; u < 8; ++u) {
      acc = wmma_f32(a[u], bf[u], acc);
      ns += a[u].x + a[u].y;
    }
  }

  // rowsum over both lane halves: lanes l and l^16 both end with nn of row n0+(l&15)
  ns += __shfl_xor(ns, 16, 32);

  float* op = out + ((size_t)(b * NN + n0 + hi * 8)) * C + c0 + lo;
#pragma unroll
  for (int j = 0; j < 8; ++j) {
    const float nn = __shfl(ns, j + hi * 8, 32);        // nn of row n0 + j + hi*8
    const float rn = 1.f / fmaxf(nn, 1e-5f);
    float* p = op + (size_t)j * C;
    *p = *p + acc[j] * rn;                              // hs (from kernel 1) + hr
  }
}

// ---------------------------------------------------------------------------
extern "C" void kernel_launch(void* const* d_in, const int* in_sizes, int n_in,
                              void* d_out, int out_size, void* d_ws, size_t ws_size,
                              hipStream_t stream) {
  const float* x     = (const float*)d_in[0];
  const float* adj   = (const float*)d_in[1];
  const float* gamma = (const float*)d_in[2];
  const float* beta  = (const float*)d_in[3];
  const float* W_hs  = (const float*)d_in[4];
  const float* b_hs  = (const float*)d_in[5];
  const float* W_hr  = (const float*)d_in[6];
  const float* b_hr  = (const float*)d_in[7];
  float* out  = (float*)d_out;
  float* hr_t = (float*)d_ws;              // [8,1024,128,4] f32 = 16 MB scratch

  ln_gemm_kernel<<<dim3(NB * NN / 16), dim3(256), 0, stream>>>(
      x, gamma, beta, W_hs, b_hs, W_hr, b_hr, out, hr_t);
  einsum_kernel<<<dim3(NN / 16, NB), dim3(256), 0, stream>>>(adj, hr_t, out);
}